// FeaturePropagation_38173669326853
// MI455X (gfx1250) — compile-verified
//
#include <hip/hip_runtime.h>
#include <math.h>

typedef __attribute__((ext_vector_type(2))) float v2f;
typedef __attribute__((ext_vector_type(8))) float v8f;

#define DD 128
#define SA_STRIDE 132   // pad 128-float rows to kill 64-bank conflicts

// ---------------------------------------------------------------------------
// Kernel 1: G = centroid_features[1024,128] @ W1[0:128,:]  (f32 WMMA 16x16x4)
// Also zeroes BN accumulators (block 0).
// ---------------------------------------------------------------------------
__global__ __launch_bounds__(256) void gemmG_kernel(
    const float* __restrict__ feat, const float* __restrict__ W1,
    float* __restrict__ G, float* __restrict__ sSum, float* __restrict__ sSq)
{
    __shared__ float sA[64 * SA_STRIDE];
    const int tid  = threadIdx.x;
    const int base = blockIdx.x * 64;

    if (blockIdx.x == 0 && tid < DD) { sSum[tid] = 0.f; sSq[tid] = 0.f; }

    for (int e = tid; e < 64 * DD; e += 256) {
        int r = e >> 7, d = e & 127;
        sA[r * SA_STRIDE + d] = feat[(base + r) * DD + d];
    }
    __syncthreads();

    const int wave = tid >> 5, lane = tid & 31;
    const int hl = lane >> 4, l16 = lane & 15;
    const int c = wave;                       // column tile 0..7

    // B fragments for this wave's column tile, held in registers (W1a rows 0..127)
    v2f bf[32];
#pragma unroll
    for (int kk = 0; kk < 32; ++kk) {
        int k0 = kk * 4 + 2 * hl;
        bf[kk].x = W1[(k0 + 0) * DD + c * 16 + l16];
        bf[kk].y = W1[(k0 + 1) * DD + c * 16 + l16];
    }

    v8f acc[4];
#pragma unroll
    for (int rt = 0; rt < 4; ++rt) acc[rt] = (v8f){0,0,0,0,0,0,0,0};

#pragma unroll
    for (int kk = 0; kk < 32; ++kk) {
#pragma unroll
        for (int rt = 0; rt < 4; ++rt) {
            v2f a;
            int ar = rt * 16 + l16;
            a.x = sA[ar * SA_STRIDE + kk * 4 + 2 * hl];
            a.y = sA[ar * SA_STRIDE + kk * 4 + 2 * hl + 1];
            acc[rt] = __builtin_amdgcn_wmma_f32_16x16x4_f32(
                false, a, false, bf[kk], (short)0, acc[rt], false, false);
        }
    }

#pragma unroll
    for (int rt = 0; rt < 4; ++rt)
#pragma unroll
        for (int r = 0; r < 8; ++r) {
            int row = base + rt * 16 + r + 8 * hl;
            G[row * DD + c * 16 + l16] = acc[rt][r];
        }
}

// ---------------------------------------------------------------------------
// Kernel 2: per-vertex top-3 centroids (by d^2) + normalized 1/d^2 weights
// ---------------------------------------------------------------------------
__global__ __launch_bounds__(256) void topk_kernel(
    const float* __restrict__ verts, const float* __restrict__ cents,
    int* __restrict__ tIdx, float* __restrict__ tW, int N, int M)
{
    __shared__ float cx[1024], cy[1024], cz[1024];
    const int tid = threadIdx.x;
    for (int m = tid; m < M; m += 256) {
        cx[m] = cents[m * 3 + 0];
        cy[m] = cents[m * 3 + 1];
        cz[m] = cents[m * 3 + 2];
    }
    __syncthreads();

    const int i = blockIdx.x * 256 + tid;
    if (i >= N) return;

    const float vx = verts[i * 3 + 0], vy = verts[i * 3 + 1], vz = verts[i * 3 + 2];
    float d0 = 3.0e38f, d1 = 3.0e38f, d2v = 3.0e38f;
    int   i0 = 0, i1 = 0, i2 = 0;

    for (int m = 0; m < M; ++m) {
        float dx = vx - cx[m], dy = vy - cy[m], dz = vz - cz[m];
        float q = dx * dx + dy * dy + dz * dz;
        if (q < d0)       { d2v = d1; i2 = i1; d1 = d0; i1 = i0; d0 = q; i0 = m; }
        else if (q < d1)  { d2v = d1; i2 = i1; d1 = q;  i1 = m; }
        else if (q < d2v) { d2v = q;  i2 = m; }
    }

    float w0, w1, w2;
    if (d0 <= 0.f) { w0 = 1.f; w1 = 0.f; w2 = 0.f; }   // exact-match special case
    else {
        w0 = 1.f / d0; w1 = 1.f / d1; w2 = 1.f / d2v;  // p=2: 1/dist^2 == 1/d2
        float s = w0 + w1 + w2;
        w0 /= s; w1 /= s; w2 /= s;
    }
    tIdx[i * 3 + 0] = i0; tIdx[i * 3 + 1] = i1; tIdx[i * 3 + 2] = i2;
    tW[i * 3 + 0] = w0;  tW[i * 3 + 1] = w1;  tW[i * 3 + 2] = w2;
}

// ---------------------------------------------------------------------------
// Kernel 3: BN statistics of h = sum_k w_k*G[idx_k] + b1 (h recomputed, not stored)
// ---------------------------------------------------------------------------
__global__ __launch_bounds__(256) void stats_kernel(
    const int* __restrict__ tIdx, const float* __restrict__ tW,
    const float* __restrict__ G, const float* __restrict__ b1,
    float* __restrict__ sSum, float* __restrict__ sSq, int N)
{
    const int tid = threadIdx.x;
    const int d = tid & 127, grp = tid >> 7;
    const int base = blockIdx.x * 32 + grp * 16;
    const float bias = b1[d];
    float s = 0.f, s2 = 0.f;
    for (int v = 0; v < 16; ++v) {
        int i = base + v;
        if (i < N) {
            int j = i * 3;
            float h = tW[j + 0] * G[tIdx[j + 0] * DD + d]
                    + tW[j + 1] * G[tIdx[j + 1] * DD + d]
                    + tW[j + 2] * G[tIdx[j + 2] * DD + d] + bias;
            s += h; s2 += h * h;
        }
    }
    atomicAdd(&sSum[d], s);
    atomicAdd(&sSq[d], s2);
}

// ---------------------------------------------------------------------------
// Kernel 4: fold BN + b1 into per-column affine: scale, shift
// ---------------------------------------------------------------------------
__global__ void finalize_kernel(
    const float* __restrict__ sSum, const float* __restrict__ sSq,
    const float* __restrict__ gamma, const float* __restrict__ beta,
    const float* __restrict__ b1,
    float* __restrict__ scale, float* __restrict__ shiftv, int N)
{
    int d = threadIdx.x;
    float inv_n = 1.f / (float)N;
    float mu  = sSum[d] * inv_n;
    float var = sSq[d] * inv_n - mu * mu;
    float inv = rsqrtf(var + 1e-5f);
    float sc  = gamma[d] * inv;
    scale[d]  = sc;
    shiftv[d] = beta[d] + (b1[d] - mu) * sc;   // b1 folded into shift
}

// ---------------------------------------------------------------------------
// Kernel 5: out = relu(g*scale + shift) @ W2 + b2   (f32 WMMA 16x16x4)
// ---------------------------------------------------------------------------
__global__ __launch_bounds__(256) void gemm2_kernel(
    const int* __restrict__ tIdx, const float* __restrict__ tW,
    const float* __restrict__ G, const float* __restrict__ scale,
    const float* __restrict__ shiftv, const float* __restrict__ W2,
    const float* __restrict__ b2, float* __restrict__ out, int N)
{
    __shared__ float sA[64 * SA_STRIDE];
    __shared__ int   sI[64 * 3];
    __shared__ float sWt[64 * 3];

    const int tid  = threadIdx.x;
    const int base = blockIdx.x * 64;

    if (tid < 64 * 3) {
        int r = tid / 3;
        int row = base + r;
        if (row < N) { sI[tid] = tIdx[row * 3 + tid % 3]; sWt[tid] = tW[row * 3 + tid % 3]; }
        else         { sI[tid] = 0;                       sWt[tid] = 0.f; }
    }
    __syncthreads();

    for (int e = tid; e < 64 * DD; e += 256) {
        int r = e >> 7, d = e & 127;
        int row = base + r;
        float h = 0.f;
        if (row < N) {
            float g = sWt[r * 3 + 0] * G[sI[r * 3 + 0] * DD + d]
                    + sWt[r * 3 + 1] * G[sI[r * 3 + 1] * DD + d]
                    + sWt[r * 3 + 2] * G[sI[r * 3 + 2] * DD + d];
            h = g * scale[d] + shiftv[d];
            h = h > 0.f ? h : 0.f;               // ReLU
        }
        sA[r * SA_STRIDE + d] = h;
    }
    __syncthreads();

    const int wave = tid >> 5, lane = tid & 31;
    const int hl = lane >> 4, l16 = lane & 15;
    const int c = wave;

    v2f bf[32];
#pragma unroll
    for (int kk = 0; kk < 32; ++kk) {
        int k0 = kk * 4 + 2 * hl;
        bf[kk].x = W2[(k0 + 0) * DD + c * 16 + l16];
        bf[kk].y = W2[(k0 + 1) * DD + c * 16 + l16];
    }

    v8f acc[4];
#pragma unroll
    for (int rt = 0; rt < 4; ++rt) acc[rt] = (v8f){0,0,0,0,0,0,0,0};

#pragma unroll
    for (int kk = 0; kk < 32; ++kk) {
#pragma unroll
        for (int rt = 0; rt < 4; ++rt) {
            v2f a;
            int ar = rt * 16 + l16;
            a.x = sA[ar * SA_STRIDE + kk * 4 + 2 * hl];
            a.y = sA[ar * SA_STRIDE + kk * 4 + 2 * hl + 1];
            acc[rt] = __builtin_amdgcn_wmma_f32_16x16x4_f32(
                false, a, false, bf[kk], (short)0, acc[rt], false, false);
        }
    }

    float bb = b2[c * 16 + l16];
#pragma unroll
    for (int rt = 0; rt < 4; ++rt)
#pragma unroll
        for (int r = 0; r < 8; ++r) {
            int row = base + rt * 16 + r + 8 * hl;
            if (row < N) out[row * DD + c * 16 + l16] = acc[rt][r] + bb;
        }
}

// ---------------------------------------------------------------------------
extern "C" void kernel_launch(void* const* d_in, const int* in_sizes, int n_in,
                              void* d_out, int out_size, void* d_ws, size_t ws_size,
                              hipStream_t stream)
{
    const float* verts = (const float*)d_in[0];
    const float* cents = (const float*)d_in[1];
    const float* feat  = (const float*)d_in[2];
    const float* W1    = (const float*)d_in[3];
    const float* b1    = (const float*)d_in[4];
    const float* gamma = (const float*)d_in[5];
    const float* beta  = (const float*)d_in[6];
    const float* W2    = (const float*)d_in[7];
    const float* b2    = (const float*)d_in[8];
    float* out = (float*)d_out;

    const int N = in_sizes[0] / 3;   // 100000
    const int M = in_sizes[1] / 3;   // 1024

    char* ws = (char*)d_ws;
    auto au = [](size_t x) { return (x + 255) & ~(size_t)255; };
    size_t o = 0;
    int*   tIdx   = (int*)(ws + o);   o += au((size_t)N * 3 * sizeof(int));
    float* tW     = (float*)(ws + o); o += au((size_t)N * 3 * sizeof(float));
    float* G      = (float*)(ws + o); o += au((size_t)M * DD * sizeof(float));
    float* sSum   = (float*)(ws + o); o += au(DD * sizeof(float));
    float* sSq    = (float*)(ws + o); o += au(DD * sizeof(float));
    float* scale  = (float*)(ws + o); o += au(DD * sizeof(float));
    float* shiftv = (float*)(ws + o); o += au(DD * sizeof(float));

    gemmG_kernel  <<<M / 64,          256, 0, stream>>>(feat, W1, G, sSum, sSq);
    topk_kernel   <<<(N + 255) / 256, 256, 0, stream>>>(verts, cents, tIdx, tW, N, M);
    stats_kernel  <<<(N + 31) / 32,   256, 0, stream>>>(tIdx, tW, G, b1, sSum, sSq, N);
    finalize_kernel<<<1,              128, 0, stream>>>(sSum, sSq, gamma, beta, b1, scale, shiftv, N);
    gemm2_kernel  <<<(N + 63) / 64,   256, 0, stream>>>(tIdx, tW, G, scale, shiftv, W2, b2, out, N);
}